// OFTRotationModule_85005992722464
// MI455X (gfx1250) — compile-verified
//
#include <hip/hip_runtime.h>
#include <hip/hip_bf16.h>

typedef __attribute__((ext_vector_type(2))) float v2f;
typedef __attribute__((ext_vector_type(8))) float v8f;

#define RBLOCKS   64
#define BK        64          // block size (K and N of each rotation)
#define NELEM     2016        // strict upper triangle of 64x64
#define D         4096        // RBLOCKS*BK, feature dim of x
#define LSTR      68          // LDS row stride in floats (bank-conflict-free, 16B aligned)
#define TOK_PER_WAVE 32
#define WAVES     8
#define TOK_PER_WG (TOK_PER_WAVE * WAVES)   // 256

// ---------------------------------------------------------------------------
// Phase 1: R = I + 2Q + 2Q^2 + 2Q^3 + 2Q^4 from packed skew weights.
// One workgroup per r-block. Writes R TRANSPOSED (Rt[c][k] = R[k][c]) so the
// phase-2 WMMA B-fragments (consecutive K per lane, fixed column) are
// contiguous ds_load_b64's.
// ---------------------------------------------------------------------------
__global__ __launch_bounds__(256)
void oft_build_rot(const float* __restrict__ w, float* __restrict__ Rt)
{
    __shared__ float Q [BK * 65];
    __shared__ float P0[BK * 65];
    __shared__ float P1[BK * 65];

    const int r   = blockIdx.x;
    const int tid = threadIdx.x;
    const float* wr = w + r * NELEM;

    // Build skew-symmetric Q (and first power P0 = Q)
    for (int idx = tid; idx < BK * BK; idx += 256) {
        int i = idx >> 6, j = idx & 63;
        float v;
        if (i < j)      v =  wr[i * (127 - i) / 2 + (j - i - 1)];
        else if (i > j) v = -wr[j * (127 - j) / 2 + (i - j - 1)];
        else            v = 0.0f;
        Q [i * 65 + j] = v;
        P0[i * 65 + j] = v;
    }
    __syncthreads();

    // Accumulate R in registers: 16 elements per thread, idx = tid + 256*q
    float acc[16];
#pragma unroll
    for (int q = 0; q < 16; ++q) {
        int idx = tid + 256 * q;
        int i = idx >> 6, j = idx & 63;
        acc[q] = (i == j ? 1.0f : 0.0f) + 2.0f * Q[i * 65 + j];
    }

    float* src = P0;
    float* dst = P1;
    for (int p = 2; p <= 4; ++p) {          // Q^2, Q^3, Q^4
#pragma unroll
        for (int q = 0; q < 16; ++q) {
            int idx = tid + 256 * q;
            int i = idx >> 6, j = idx & 63;
            float s = 0.0f;
            for (int k = 0; k < BK; ++k)
                s = fmaf(src[i * 65 + k], Q[k * 65 + j], s);
            dst[i * 65 + j] = s;
            acc[q] += 2.0f * s;
        }
        __syncthreads();
        float* t = src; src = dst; dst = t;
    }

    // Write transposed: Rt[r][c*64 + k] = R[k][c]
    float* out = Rt + r * (BK * BK);
#pragma unroll
    for (int q = 0; q < 16; ++q) {
        int idx = tid + 256 * q;
        int i = idx >> 6, j = idx & 63;     // R[i][j]
        out[j * BK + i] = acc[q];
    }
}

// ---------------------------------------------------------------------------
// Phase 2: y[t, r*64 + c] = sum_k x[t, r*64 + k] * R[r][k][c]
// Memory-bound fp32 stream (512 MB total); V_WMMA_F32_16X16X4_F32 compute.
// Workgroup = 8 waves; wave = 32-token x 64-col tile (2x4 16x16 C tiles).
// ---------------------------------------------------------------------------
__global__ __launch_bounds__(256)
void oft_apply(const float* __restrict__ x, const float* __restrict__ Rt,
               float* __restrict__ y, int tgCount)
{
    __shared__ float sRt[BK * LSTR];                        // Rt for this r-block
    __shared__ float sX [WAVES * TOK_PER_WAVE * LSTR];      // wave-private tiles

    const int r    = blockIdx.y;
    const int tid  = threadIdx.x;
    const int wave = tid >> 5;
    const int lane = tid & 31;
    const int laneM = lane & 15;
    const int kHi   = (lane >> 4) << 1;     // 0 for lanes 0-15, 2 for 16-31
    const int mHi   = (lane >> 4) << 3;     // 0 / 8  (C/D row offset)

    // Cooperative load of Rt[r] (64x64 f32) into LDS, stride-68 rows.
    {
        const float* src = Rt + (size_t)r * (BK * BK);
        for (int i = tid; i < BK * 16; i += 256) {          // 16 float4-chunks/row
            int row = i >> 4, ch = i & 15;
            float4 v = *(const float4*)(src + row * BK + ch * 4);
            *(float4*)&sRt[row * LSTR + ch * 4] = v;
        }
    }
    __syncthreads();

    float* sXw = &sX[wave * TOK_PER_WAVE * LSTR];

    for (int tg = blockIdx.x; tg < tgCount; tg += gridDim.x) {
        const size_t tokBase = (size_t)tg * TOK_PER_WG + wave * TOK_PER_WAVE;

        // Stage 32 rows x 64 f32 (wave-private): coalesced b128 global loads,
        // 16B-aligned b128 LDS stores. Per-wave DS ordering => no barrier.
        const float* gx = x + tokBase * D + r * BK;
#pragma unroll
        for (int it = 0; it < 16; ++it) {
            int i = it * 32 + lane;
            int row = i >> 4, ch = i & 15;
            float4 v = *(const float4*)(gx + (size_t)row * D + ch * 4);
            *(float4*)&sXw[row * LSTR + ch * 4] = v;
        }

        v8f c[2][4] = {};

        // K loop: 16 steps of 16x16x4 f32 WMMA, 2 M-tiles x 4 N-tiles
        for (int s = 0; s < 16; ++s) {
            const int kOff = 4 * s + kHi;
            v2f a0 = *(const v2f*)&sXw[(laneM +  0) * LSTR + kOff];
            v2f a1 = *(const v2f*)&sXw[(laneM + 16) * LSTR + kOff];
#pragma unroll
            for (int n = 0; n < 4; ++n) {
                v2f b = *(const v2f*)&sRt[(n * 16 + laneM) * LSTR + kOff];
                c[0][n] = __builtin_amdgcn_wmma_f32_16x16x4_f32(
                    false, a0, false, b, (short)0, c[0][n], false, false);
                c[1][n] = __builtin_amdgcn_wmma_f32_16x16x4_f32(
                    false, a1, false, b, (short)0, c[1][n], false, false);
            }
        }

        // Store: VGPR v of a 16x16 f32 D tile holds (M=v | M=v+8, N=lane%16)
        float* gy = y + tokBase * D + r * BK;
#pragma unroll
        for (int m = 0; m < 2; ++m)
#pragma unroll
            for (int n = 0; n < 4; ++n)
#pragma unroll
                for (int v = 0; v < 8; ++v) {
                    int row = m * 16 + v + mHi;
                    gy[(size_t)row * D + n * 16 + laneM] = c[m][n][v];
                }
    }
}

extern "C" void kernel_launch(void* const* d_in, const int* in_sizes, int n_in,
                              void* d_out, int out_size, void* d_ws, size_t ws_size,
                              hipStream_t stream)
{
    const float* x  = (const float*)d_in[0];
    const float* w  = (const float*)d_in[1];
    float*       y  = (float*)d_out;
    float*       Rt = (float*)d_ws;                 // 64*64*64*4 = 1 MB scratch

    const int tokens  = in_sizes[0] / D;            // 4*4096 = 16384
    const int tgCount = tokens / TOK_PER_WG;        // 64

    oft_build_rot<<<dim3(RBLOCKS), dim3(256), 0, stream>>>(w, Rt);
    oft_apply<<<dim3(32, RBLOCKS), dim3(256), 0, stream>>>(x, Rt, y, tgCount);
}